// DRRGHead_76124000354366
// MI455X (gfx1250) — compile-verified
//
#include <hip/hip_runtime.h>
#include <hip/hip_bf16.h>

typedef float v2f __attribute__((ext_vector_type(2)));
typedef float v8f __attribute__((ext_vector_type(8)));

#define GCNT  512   // graphs
#define NNODE 41    // nodes per graph
#define MPAD  48    // padded rows (3 x 16 WMMA M-tiles)
#define F0    576   // input feature dim

// D = A(16x4) * B(4x16) + C, all f32, wave32. ISA §7.12.2 layout:
// A frag (per lane): elem0 = (M=lane&15, K=k+2*(lane>>4)), elem1 = K+1
// B frag (per lane): elem0 = (K=k+2*(lane>>4), N=lane&15), elem1 = K+1
// C/D: vgpr r holds (M = r + 8*(lane>>4), N = lane&15)
__device__ __forceinline__ v8f wmma4(v2f a, v2f b, v8f c) {
    return __builtin_amdgcn_wmma_f32_16x16x4_f32(
        /*neg_a=*/false, a, /*neg_b=*/false, b,
        /*c_mod=*/(short)0, c, /*reuse_a=*/false, /*reuse_b=*/false);
}

// ---------------------------------------------------------------- 1x1 conv
// pred_maps[b,o,h,w] = sum_c inputs[b,c,h,w] * conv_w[o,c] + conv_b[o]
// 160 MB of traffic @ 2.5 FLOP/byte -> pure HBM streaming, VALU only.
__global__ void conv1x1_kernel(const float* __restrict__ in,
                               const float* __restrict__ w,
                               const float* __restrict__ b,
                               float* __restrict__ out) {
    __shared__ float wls[6 * 32];
    __shared__ float bls[6];
    if (threadIdx.x < 192) wls[threadIdx.x] = w[threadIdx.x];
    if (threadIdx.x < 6)   bls[threadIdx.x] = b[threadIdx.x];
    __syncthreads();
    const int HW = 512 * 512;
    int p  = blockIdx.x * blockDim.x + threadIdx.x;   // 4*HW pixels
    int bb = p / HW, hw = p % HW;
    const float* ip = in + (size_t)bb * 32 * HW + hw;
    float o[6];
#pragma unroll
    for (int j = 0; j < 6; ++j) o[j] = bls[j];
#pragma unroll
    for (int c = 0; c < 32; ++c) {
        float v = ip[(size_t)c * HW];
#pragma unroll
        for (int j = 0; j < 6; ++j) o[j] += v * wls[j * 32 + c];
    }
    float* op = out + (size_t)bb * 6 * HW + hw;
#pragma unroll
    for (int j = 0; j < 6; ++j) op[(size_t)j * HW] = o[j];
}

// ------------------------------------------------------------- BatchNorm1d
__global__ void bn_stats_kernel(const float* __restrict__ x,
                                float* __restrict__ mean,
                                float* __restrict__ rstd) {
    const int f = blockIdx.x;              // feature 0..575
    const int R = GCNT * NNODE;            // 20992 rows
    float s = 0.f, s2 = 0.f;
    for (int r = threadIdx.x; r < R; r += blockDim.x) {
        float v = x[(size_t)r * F0 + f];
        s += v; s2 += v * v;
    }
    __shared__ float red0[256], red1[256];
    red0[threadIdx.x] = s; red1[threadIdx.x] = s2;
    __syncthreads();
    for (int wdt = 128; wdt > 0; wdt >>= 1) {
        if ((int)threadIdx.x < wdt) {
            red0[threadIdx.x] += red0[threadIdx.x + wdt];
            red1[threadIdx.x] += red1[threadIdx.x + wdt];
        }
        __syncthreads();
    }
    if (threadIdx.x == 0) {
        float m = red0[0] / (float)R;
        float v = red1[0] / (float)R - m * m;
        mean[f] = m;
        rstd[f] = rsqrtf(v + 1e-5f);
    }
}

// normalize + write 48-row padded layout (pad rows = 0)
__global__ void bn_apply_kernel(const float* __restrict__ x,
                                const float* __restrict__ mean,
                                const float* __restrict__ rstd,
                                float* __restrict__ y) {
    size_t idx = (size_t)blockIdx.x * blockDim.x + threadIdx.x; // G*48*576
    int f = (int)(idx % F0);
    int r = (int)((idx / F0) % MPAD);
    int g = (int)(idx / ((size_t)F0 * MPAD));
    float v = 0.f;
    if (r < NNODE)
        v = (x[((size_t)g * NNODE + r) * F0 + f] - mean[f]) * rstd[f];
    y[idx] = v;
}

// ----------------------------------------------- agg = A(41x41) @ x(48xF)
// One block (8 waves) per graph; adjacency zero-padded to 48x48 in LDS.
// Each wave owns one N-tile column and all 3 M-tiles: per K-step the two
// strided x loads feed 3 independent v_wmma (no D->A/B hazard stalls).
template <int F>
__global__ void agg_kernel(const float* __restrict__ A,
                           const float* __restrict__ x,
                           float* __restrict__ agg) {
    const int g = blockIdx.x;
    __shared__ float Als[MPAD * MPAD];     // 9 KB
    for (int i = threadIdx.x; i < MPAD * MPAD; i += blockDim.x) Als[i] = 0.f;
    __syncthreads();
    for (int i = threadIdx.x; i < NNODE * NNODE; i += blockDim.x) {
        int r = i / NNODE, c = i % NNODE;
        Als[r * MPAD + c] = A[((size_t)g * NNODE + r) * NNODE + c];
    }
    __syncthreads();

    const int lane = threadIdx.x & 31;
    const int wave = threadIdx.x >> 5;     // 0..7
    const int half = lane >> 4;            // K-half select
    const int l16  = lane & 15;
    const float* xg   = x   + (size_t)g * MPAD * F;
    float*       aggg = agg + (size_t)g * MPAD * F;

    for (int nt = wave; nt < F / 16; nt += 8) {   // wave-uniform trip count
        const int col = nt * 16 + l16;
        v8f acc0 = {}, acc1 = {}, acc2 = {};
        const float* als  = &Als[l16 * MPAD];     // M-row base (mt=0)
        const float* xcol = xg + col;
#pragma unroll 4
        for (int k = 0; k < MPAD; k += 4) {
            const int kk = k + 2 * half;
            v2f b;
            b.x = xcol[(kk)*F];                  // const stride -> imm offsets
            b.y = xcol[(kk + 1) * F];
            v2f a0 = *(const v2f*)(als + kk);             // ds_load_b64
            v2f a1 = *(const v2f*)(als + 16 * MPAD + kk);
            v2f a2 = *(const v2f*)(als + 32 * MPAD + kk);
            acc0 = wmma4(a0, b, acc0);
            acc1 = wmma4(a1, b, acc1);
            acc2 = wmma4(a2, b, acc2);
        }
        float* ocol = aggg + (8 * half) * F + col;
#pragma unroll
        for (int r = 0; r < 8; ++r) {
            ocol[(r)*F]      = acc0[r];
            ocol[(16 + r) * F] = acc1[r];
            ocol[(32 + r) * F] = acc2[r];
        }
    }
}

// -------------------------- y = relu([x | agg](48 x 2Fin) @ W(2Fin x Fout) + b)
// Same wave tiling: 1 N-tile x 3 M-tiles per wave; W loads amortized 3x.
template <int Fin, int Fout>
__global__ void dense_kernel(const float* __restrict__ x,
                             const float* __restrict__ agg,
                             const float* __restrict__ W,
                             const float* __restrict__ bias,
                             float* __restrict__ y) {
    const int g    = blockIdx.x;
    const int lane = threadIdx.x & 31;
    const int wave = threadIdx.x >> 5;
    const int half = lane >> 4;
    const int l16  = lane & 15;
    const float* xg = x   + (size_t)g * MPAD * Fin;
    const float* ag = agg + (size_t)g * MPAD * Fin;
    float*       yg = y   + (size_t)g * MPAD * Fout;

    for (int nt = wave; nt < Fout / 16; nt += 8) {
        const int col = nt * 16 + l16;
        v8f acc0 = {}, acc1 = {}, acc2 = {};
        // K segment 1: x part (W rows [0, Fin))
        {
            const float* arow = xg + l16 * Fin;   // mt=0 row; mt offsets const
            const float* wcol = W + col;
#pragma unroll 4
            for (int k = 0; k < Fin; k += 4) {
                const int kk = k + 2 * half;
                v2f b;
                b.x = wcol[(kk)*Fout];
                b.y = wcol[(kk + 1) * Fout];
                v2f a0 = *(const v2f*)(arow + kk);                // b64, 8B aligned
                v2f a1 = *(const v2f*)(arow + 16 * Fin + kk);
                v2f a2 = *(const v2f*)(arow + 32 * Fin + kk);
                acc0 = wmma4(a0, b, acc0);
                acc1 = wmma4(a1, b, acc1);
                acc2 = wmma4(a2, b, acc2);
            }
        }
        // K segment 2: agg part (W rows [Fin, 2Fin))
        {
            const float* arow = ag + l16 * Fin;
            const float* wcol = W + (size_t)Fin * Fout + col;
#pragma unroll 4
            for (int k = 0; k < Fin; k += 4) {
                const int kk = k + 2 * half;
                v2f b;
                b.x = wcol[(kk)*Fout];
                b.y = wcol[(kk + 1) * Fout];
                v2f a0 = *(const v2f*)(arow + kk);
                v2f a1 = *(const v2f*)(arow + 16 * Fin + kk);
                v2f a2 = *(const v2f*)(arow + 32 * Fin + kk);
                acc0 = wmma4(a0, b, acc0);
                acc1 = wmma4(a1, b, acc1);
                acc2 = wmma4(a2, b, acc2);
            }
        }
        const float bv = bias[col];
        float* ocol = yg + (8 * half) * Fout + col;
#pragma unroll
        for (int r = 0; r < 8; ++r) {
            float v0 = acc0[r] + bv, v1 = acc1[r] + bv, v2 = acc2[r] + bv;
            ocol[(r)*Fout]      = v0 > 0.f ? v0 : 0.f;
            ocol[(16 + r) * Fout] = v1 > 0.f ? v1 : 0.f;
            ocol[(32 + r) * Fout] = v2 > 0.f ? v2 : 0.f;
        }
    }
}

// -------------------------------------------------- classifier head (tiny)
__global__ void classifier_kernel(const float* __restrict__ x,   // G x 48 x 64
                                  const int*   __restrict__ knn, // G x 8
                                  const float* __restrict__ w1,  // 64x32
                                  const float* __restrict__ b1,
                                  const float* __restrict__ pa,  // PReLU alpha
                                  const float* __restrict__ w2,  // 32x2
                                  const float* __restrict__ b2,
                                  float* __restrict__ out) {     // (G*8) x 2
    int e = blockIdx.x * blockDim.x + threadIdx.x;
    if (e >= GCNT * 8) return;
    int g    = e >> 3;
    int node = knn[e];
    const float* row = x + ((size_t)g * MPAD + node) * 64;
    float o0 = b2[0], o1 = b2[1];
    for (int j = 0; j < 32; ++j) {
        float h = b1[j];
#pragma unroll 8
        for (int c = 0; c < 64; ++c) h += row[c] * w1[c * 32 + j];
        h = (h >= 0.f) ? h : pa[j] * h;
        o0 += h * w2[j * 2 + 0];
        o1 += h * w2[j * 2 + 1];
    }
    out[(size_t)e * 2 + 0] = o0;
    out[(size_t)e * 2 + 1] = o1;
}

extern "C" void kernel_launch(void* const* d_in, const int* in_sizes, int n_in,
                              void* d_out, int out_size, void* d_ws, size_t ws_size,
                              hipStream_t stream) {
    const float* inputs     = (const float*)d_in[0];
    const float* node_feats = (const float*)d_in[1];
    const float* A          = (const float*)d_in[2];
    const int*   knn        = (const int*)  d_in[3];
    const float* conv_w     = (const float*)d_in[4];
    const float* conv_b     = (const float*)d_in[5];
    const float* W1 = (const float*)d_in[6],  *b1 = (const float*)d_in[7];
    const float* W2 = (const float*)d_in[8],  *b2 = (const float*)d_in[9];
    const float* W3 = (const float*)d_in[10], *b3 = (const float*)d_in[11];
    const float* W4 = (const float*)d_in[12], *b4 = (const float*)d_in[13];
    const float* cls_w1  = (const float*)d_in[14];
    const float* cls_b1  = (const float*)d_in[15];
    const float* prelu_a = (const float*)d_in[16];
    const float* cls_w2  = (const float*)d_in[17];
    const float* cls_b2  = (const float*)d_in[18];

    float* out = (float*)d_out;
    float* ws  = (float*)d_ws;

    const size_t BUF = (size_t)GCNT * MPAD * F0;   // 14,155,776 floats each
    float* B0   = ws;
    float* B1   = ws + BUF;          // agg scratch
    float* B2   = ws + 2 * BUF;
    float* mean = ws + 3 * BUF;
    float* rstd = mean + F0;

    // 1x1 conv -> d_out[0 .. 4*6*512*512)
    conv1x1_kernel<<<(4 * 512 * 512) / 256, 256, 0, stream>>>(inputs, conv_w, conv_b, out);

    // BatchNorm (batch stats, biased var) -> padded x0 in B0
    bn_stats_kernel<<<F0, 256, 0, stream>>>(node_feats, mean, rstd);
    bn_apply_kernel<<<(unsigned)((BUF + 255) / 256), 256, 0, stream>>>(node_feats, mean, rstd, B0);

    // 4 GraphConv layers (agg then dense), ping-pong B0 <-> B2, agg in B1
    agg_kernel<576>        <<<GCNT, 256, 0, stream>>>(A, B0, B1);
    dense_kernel<576, 512> <<<GCNT, 256, 0, stream>>>(B0, B1, W1, b1, B2);

    agg_kernel<512>        <<<GCNT, 256, 0, stream>>>(A, B2, B1);
    dense_kernel<512, 256> <<<GCNT, 256, 0, stream>>>(B2, B1, W2, b2, B0);

    agg_kernel<256>        <<<GCNT, 256, 0, stream>>>(A, B0, B1);
    dense_kernel<256, 128> <<<GCNT, 256, 0, stream>>>(B0, B1, W3, b3, B2);

    agg_kernel<128>        <<<GCNT, 256, 0, stream>>>(A, B2, B1);
    dense_kernel<128, 64>  <<<GCNT, 256, 0, stream>>>(B2, B1, W4, b4, B0);

    // knn gather + MLP head -> d_out tail
    classifier_kernel<<<(GCNT * 8 + 255) / 256, 256, 0, stream>>>(
        B0, knn, cls_w1, cls_b1, prelu_a, cls_w2, cls_b2,
        out + (size_t)4 * 6 * 512 * 512);
}